// LiftedLoss_80745385165476
// MI455X (gfx1250) — compile-verified
//
#include <hip/hip_runtime.h>
#include <math.h>

typedef __attribute__((ext_vector_type(2))) float v2f;
typedef __attribute__((ext_vector_type(8))) float v8f;

#define NROWS 8192
#define DIM 128
#define MARGIN_F 1.0f

// ---------------------------------------------------------------------------
// Kernel 1: row squared norms. score is 4MB total -> trivial.
// ---------------------------------------------------------------------------
__global__ void mag_kernel(const float* __restrict__ score, float* __restrict__ mag) {
    int row = blockIdx.x * blockDim.x + threadIdx.x;
    if (row >= NROWS) return;
    const float4* p = (const float4*)(score + (size_t)row * DIM);
    float s = 0.0f;
#pragma unroll
    for (int k = 0; k < DIM / 4; ++k) {
        float4 v = p[k];
        s += v.x * v.x + v.y * v.y + v.z * v.z + v.w * v.w;
    }
    mag[row] = s;
}

// ---------------------------------------------------------------------------
// Fused distance-tile pass. Each block owns a 16-row i-tile; its 8 waves
// split the j dimension (wave w handles j0 = w*16, stride 128). Each wave
// computes 16x16 sim tiles with V_WMMA_F32_16X16X4_F32 (f32 precision to
// match the reference through exp()).
//
// f32 WMMA operand layout (wave32):
//   A (16x4):  lane = M (0..15 per half), halves split K: lanes<16 K={0,1},
//              lanes>=16 K={2,3}; VGPR0/1 = the K pair.
//   B (4x16):  lane = N, same K split. Since B = score^T, A and B loads use
//              the same addressing with different row bases.
//   C/D (16x16 f32): lane l, vgpr v -> M = v + 8*(l>=16), N = l%16.
//
// MODE 0: accumulate N[i] = sum_{t_j != t_i} exp(margin - dist(i,j))
// MODE 1: accumulate loss = sum_{same-class, i!=j} relu(log(N_i+N_j)+dist)^2
//         and the pair count.
// ---------------------------------------------------------------------------
template <int MODE>
__global__ __launch_bounds__(256)
void pass_kernel(const float* __restrict__ score,
                 const int* __restrict__ targets,
                 const float* __restrict__ mag,
                 float* __restrict__ Nvec,          // MODE0: out, MODE1: in
                 float* __restrict__ lossP,         // MODE1: out [gridDim.x]
                 float* __restrict__ cntP)          // MODE1: out [gridDim.x]
{
    const int i0   = blockIdx.x * 16;
    const int lane = threadIdx.x & 31;
    const int w    = threadIdx.x >> 5;   // wave id 0..7
    const int h    = lane >> 4;          // half of wave
    const int m    = lane & 15;          // position within half

    // Preload all A fragments for this i-tile (invariant over the j sweep).
    const float* arow = score + (size_t)(i0 + m) * DIM + 2 * h;
    v2f afrag[32];
#pragma unroll
    for (int kk = 0; kk < 32; ++kk)
        afrag[kk] = *(const v2f*)(arow + kk * 4);

    // Per-lane metadata for the 8 rows this lane's acc VGPRs map to.
    float mi[8], ni[8];
    int   ti[8];
#pragma unroll
    for (int v = 0; v < 8; ++v) {
        int i = i0 + v + 8 * h;
        mi[v] = mag[i];
        ti[v] = targets[i];
        ni[v] = (MODE == 1) ? Nvec[i] : 0.0f;
    }

    float accN[8];
#pragma unroll
    for (int v = 0; v < 8; ++v) accN[v] = 0.0f;
    float lossAcc = 0.0f, cntAcc = 0.0f;

    for (int j0 = w * 16; j0 < NROWS; j0 += 8 * 16) {
        const int jrow = j0 + m;
        const float* brow = score + (size_t)jrow * DIM + 2 * h;
        const float magj = mag[jrow];
        const int   tj   = targets[jrow];
        const float nj   = (MODE == 1) ? Nvec[jrow] : 0.0f;

        v8f acc = {};
#pragma unroll
        for (int kk = 0; kk < 32; ++kk) {
            v2f b = *(const v2f*)(brow + kk * 4);
            acc = __builtin_amdgcn_wmma_f32_16x16x4_f32(
                false, afrag[kk], false, b, (short)0, acc, false, false);
        }

#pragma unroll
        for (int v = 0; v < 8; ++v) {
            float d2   = mi[v] + magj - 2.0f * acc[v];
            float dist = sqrtf(fmaxf(d2, 0.0f));
            if (MODE == 0) {
                if (tj != ti[v]) accN[v] += expf(MARGIN_F - dist);
            } else {
                int  i    = i0 + v + 8 * h;
                bool same = (tj == ti[v]) && (i != jrow);
                if (same) {
                    float ln = logf(ni[v] + nj) + dist;
                    float r  = fmaxf(ln, 0.0f);
                    lossAcc += r * r;
                    cntAcc  += 1.0f;
                }
            }
        }
    }

    __shared__ float red0[8][16];
    __shared__ float red1[8][2];

    if (MODE == 0) {
        // Reduce each row's partial over the 16 lanes of its half, then
        // combine the 8 waves' j-partials through LDS. No atomics.
#pragma unroll
        for (int v = 0; v < 8; ++v) {
            float s = accN[v];
#pragma unroll
            for (int off = 1; off < 16; off <<= 1)
                s += __shfl_xor(s, off, 32);
            if (m == 0) red0[w][v + 8 * h] = s;   // lanes 0 and 16 write
        }
        __syncthreads();
        if (threadIdx.x < 16) {
            float s = 0.0f;
#pragma unroll
            for (int ww = 0; ww < 8; ++ww) s += red0[ww][threadIdx.x];
            Nvec[i0 + threadIdx.x] = s;
        }
    } else {
        // Full-wave reduce of scalar loss/count, then across waves via LDS.
#pragma unroll
        for (int off = 1; off < 32; off <<= 1) {
            lossAcc += __shfl_xor(lossAcc, off, 32);
            cntAcc  += __shfl_xor(cntAcc, off, 32);
        }
        if (lane == 0) { red1[w][0] = lossAcc; red1[w][1] = cntAcc; }
        __syncthreads();
        if (threadIdx.x == 0) {
            float l = 0.0f, c = 0.0f;
#pragma unroll
            for (int ww = 0; ww < 8; ++ww) { l += red1[ww][0]; c += red1[ww][1]; }
            lossP[blockIdx.x] = l;
            cntP[blockIdx.x]  = c;
        }
    }
}

// ---------------------------------------------------------------------------
// Kernel 4: reduce 512 per-block partials, write loss / (4 * counter).
// ---------------------------------------------------------------------------
__global__ void finalize_kernel(const float* __restrict__ lossP,
                                const float* __restrict__ cntP,
                                float* __restrict__ out)
{
    __shared__ float sl[256];
    __shared__ float sc[256];
    int t = threadIdx.x;
    sl[t] = lossP[t] + lossP[t + 256];
    sc[t] = cntP[t] + cntP[t + 256];
    __syncthreads();
    for (int s = 128; s > 0; s >>= 1) {
        if (t < s) { sl[t] += sl[t + s]; sc[t] += sc[t + s]; }
        __syncthreads();
    }
    if (t == 0) out[0] = sl[0] / (4.0f * sc[0]);
}

// ---------------------------------------------------------------------------
extern "C" void kernel_launch(void* const* d_in, const int* in_sizes, int n_in,
                              void* d_out, int out_size, void* d_ws, size_t ws_size,
                              hipStream_t stream) {
    const float* score   = (const float*)d_in[0];
    const int*   targets = (const int*)d_in[1];

    float* ws    = (float*)d_ws;
    float* mag   = ws;                 // 8192
    float* Nvec  = ws + NROWS;         // 8192
    float* lossP = ws + 2 * NROWS;     // 512
    float* cntP  = ws + 2 * NROWS + 512; // 512
    float* out   = (float*)d_out;

    mag_kernel<<<NROWS / 256, 256, 0, stream>>>(score, mag);
    pass_kernel<0><<<NROWS / 16, 256, 0, stream>>>(score, targets, mag, Nvec,
                                                   nullptr, nullptr);
    pass_kernel<1><<<NROWS / 16, 256, 0, stream>>>(score, targets, mag, Nvec,
                                                   lossP, cntP);
    finalize_kernel<<<1, 256, 0, stream>>>(lossP, cntP, out);
}